// RoIAlign_8684423872826
// MI455X (gfx1250) — compile-verified
//
#include <hip/hip_runtime.h>

typedef float v2f __attribute__((ext_vector_type(2)));
typedef float v8f __attribute__((ext_vector_type(8)));

constexpr int   C_CH  = 256;
constexpr int   H_IN  = 100;
constexpr int   W_IN  = 100;
constexpr int   R_ROI = 512;
constexpr int   OH    = 7;
constexpr int   OW    = 7;
constexpr float SCALE = 0.0625f;

__global__ __launch_bounds__(256) void roialign_wmma_kernel(
    const float* __restrict__ feat,   // (2, 256, 100, 100)
    const float* __restrict__ rois,   // (512, 5): b, x1, y1, x2, y2
    float* __restrict__ out)          // (512, 256, 7, 7)
{
    const int bin = blockIdx.x;            // 0..48
    const int r   = blockIdx.y;            // 0..511
    const int ph  = bin / OW;
    const int pw  = bin - ph * OW;

    const float* roi = rois + r * 5;
    const int   b  = (int)roi[0];
    const float sx = roi[1] * SCALE - 0.5f;
    const float sy = roi[2] * SCALE - 0.5f;
    const float ex = roi[3] * SCALE - 0.5f;
    const float ey = roi[4] * SCALE - 0.5f;
    const float bh = (ey - sy) * (1.0f / OH);
    const float bw = (ex - sx) * (1.0f / OW);

    const int tid  = threadIdx.x;
    const int lane = tid & 31;
    const int wave = tid >> 5;
    const int m    = lane & 15;            // channel row inside 16-wide tile
    const int half = lane >> 4;            // 0 -> iy=0 (K0,K1); 1 -> iy=1 (K2,K3)

    // ---- y sample for this half (shared by both x samples of this lane) ----
    const float yy = sy + ((float)ph + (half ? 0.75f : 0.25f)) * bh;
    const float vy = (yy >= -1.0f && yy <= (float)H_IN) ? 1.0f : 0.0f;
    float yc = fminf(fmaxf(yy, 0.0f), (float)(H_IN - 1));
    float yb = fminf(floorf(yc), (float)(H_IN - 2));
    const int   y0  = (int)yb;
    const float wy1 = yc - yb;
    const float wy0 = 1.0f - wy1;

    // ---- two x samples (ix = 0, 1) ----
    const float xxa = sx + ((float)pw + 0.25f) * bw;
    const float xxb = sx + ((float)pw + 0.75f) * bw;
    const float vma = vy * ((xxa >= -1.0f && xxa <= (float)W_IN) ? 1.0f : 0.0f);
    const float vmb = vy * ((xxb >= -1.0f && xxb <= (float)W_IN) ? 1.0f : 0.0f);

    float xca = fminf(fmaxf(xxa, 0.0f), (float)(W_IN - 1));
    float xba = fminf(floorf(xca), (float)(W_IN - 2));
    const int   x0a  = (int)xba;
    const float wx1a = xca - xba;
    const float wx0a = 1.0f - wx1a;

    float xcb = fminf(fmaxf(xxb, 0.0f), (float)(W_IN - 1));
    float xbb = fminf(floorf(xcb), (float)(W_IN - 2));
    const int   x0b  = (int)xbb;
    const float wx1b = xcb - xbb;
    const float wx0b = 1.0f - wx1b;

    // B matrix = all 0.25 (folds the /(S*S) divide); layout-independent.
    v2f Bm;
    Bm[0] = 0.25f;
    Bm[1] = 0.25f;

    // 8 waves x 16 channels x 2 iterations = 256 channels
    for (int it = 0; it < 2; ++it) {
        const int c = wave * 16 + it * 128 + m;
        const float* row0 = feat + (((size_t)b * C_CH + c) * H_IN + y0) * W_IN;
        const float* row1 = row0 + W_IN;

        // bilinear taps (all addresses in-bounds by construction)
        const float f00a = row0[x0a], f01a = row0[x0a + 1];
        const float f10a = row1[x0a], f11a = row1[x0a + 1];
        const float f00b = row0[x0b], f01b = row0[x0b + 1];
        const float f10b = row1[x0b], f11b = row1[x0b + 1];

        const float pa = vma * (wy0 * (wx0a * f00a + wx1a * f01a) +
                                wy1 * (wx0a * f10a + wx1a * f11a));
        const float pb = vmb * (wy0 * (wx0b * f00b + wx1b * f01b) +
                                wy1 * (wx0b * f10b + wx1b * f11b));

        // A-matrix (16x4 f32): lane<16 -> row m, K0/K1; lane>=16 -> row m, K2/K3
        v2f Am;
        Am[0] = pa;
        Am[1] = pb;

        v8f Cm = {};
        // D[m, n] = 0.25 * (p0 + p1 + p2 + p3)[m]   for every column n
        v8f D = __builtin_amdgcn_wmma_f32_16x16x4_f32(
            false, Am, false, Bm, (short)0, Cm, false, false);

        // Extract: lanes 0-15 hold rows 0..7 in D[0..7]; lanes 16-31 rows 8..15.
        const int idx = lane & 7;
        const float a0 = (idx & 1) ? D[1] : D[0];
        const float a1 = (idx & 1) ? D[3] : D[2];
        const float a2 = (idx & 1) ? D[5] : D[4];
        const float a3 = (idx & 1) ? D[7] : D[6];
        const float b0 = (idx & 2) ? a1 : a0;
        const float b1 = (idx & 2) ? a3 : a2;
        const float sel = (idx & 4) ? b1 : b0;

        // route row m to lane m (m<8 from lane m, m>=8 from lane m+8)
        const int srcLane = (m < 8) ? m : (m + 8);
        const float outv = __int_as_float(
            __builtin_amdgcn_ds_bpermute(srcLane << 2, __float_as_int(sel)));

        if (lane < 16) {
            out[(((size_t)r * C_CH + c) * OH + ph) * OW + pw] = outv;
        }
    }
}

extern "C" void kernel_launch(void* const* d_in, const int* in_sizes, int n_in,
                              void* d_out, int out_size, void* d_ws, size_t ws_size,
                              hipStream_t stream) {
    const float* feat = (const float*)d_in[0];
    const float* rois = (const float*)d_in[1];
    float* out = (float*)d_out;

    dim3 grid(OH * OW, R_ROI);   // one workgroup per (bin, roi)
    dim3 block(256);             // 8 wave32 waves
    roialign_wmma_kernel<<<grid, block, 0, stream>>>(feat, rois, out);
}